// InnerProductDecoder_51539607552043
// MI455X (gfx1250) — compile-verified
//
#include <hip/hip_runtime.h>
#include <hip/hip_bf16.h>

typedef float v2f __attribute__((ext_vector_type(2)));
typedef float v8f __attribute__((ext_vector_type(8)));

#define D_FEAT 64
#define EDGES_PER_WAVE 16
#define WAVES_PER_BLOCK 8
#define EDGES_PER_BLOCK (EDGES_PER_WAVE * WAVES_PER_BLOCK)

// One wave32 computes 16 edge scores: diag( Z[rows] (16x64)  x  Z[cols]^T (64x16) )
// via 16x V_WMMA_F32_16X16X4_F32 accumulation (fp32 in/out, no precision loss).
__global__ __launch_bounds__(256) void InnerProductDecoder_wmma_kernel(
    const float* __restrict__ z,          // [N_NODES, 64] fp32
    const int*   __restrict__ edge_index, // [2, E] int32
    float*       __restrict__ out,        // [E] fp32
    int E)
{
    const int lane = threadIdx.x & 31;
    const int waveInBlock = threadIdx.x >> 5;
    const long long base =
        ((long long)blockIdx.x * WAVES_PER_BLOCK + waveInBlock) * EDGES_PER_WAVE;

    // Wave-uniform exit: EXEC stays all-ones inside (WMMA requirement).
    if (base >= E) return;

    // Lane l handles matrix row/col m = l&15; upper half-wave covers K+2..K+3.
    const int m = lane & 15;
    long long e = base + m;
    if (e >= (long long)E) e = (long long)E - 1;  // clamp (cndmask, no EXEC change)

    const int rIdx = edge_index[e];                   // row = edge_index[0][e]
    const int cIdx = edge_index[(long long)E + e];    // col = edge_index[1][e]

    const float* __restrict__ aRow = z + (long long)rIdx * D_FEAT;
    const float* __restrict__ bRow = z + (long long)cIdx * D_FEAT;
    const int kOff = (lane >> 4) * 2;   // lanes 0-15: K+0/K+1; lanes 16-31: K+2/K+3

    v8f acc = {};   // C = 0

    #pragma unroll
    for (int k0 = 0; k0 < D_FEAT; k0 += 4) {
        // A 16x4 tile: lane holds {A[m][k0+kOff], A[m][k0+kOff+1]}  (global_load_b64)
        v2f a = *(const v2f*)(aRow + k0 + kOff);
        // B 4x16 tile mirrors A with M<->N (row striped across lanes per VGPR)
        v2f b = *(const v2f*)(bRow + k0 + kOff);
        // D = A x B + C ; 8 args: (neg_a, A, neg_b, B, c_mod, C, reuse_a, reuse_b)
        acc = __builtin_amdgcn_wmma_f32_16x16x4_f32(
            false, a, false, b, (short)0, acc, false, false);
    }

    // Diagonal extraction:
    //   m in 0..7  -> lane m,     VGPR m
    //   m in 8..15 -> lane m+16,  VGPR m-8
    const int sel = lane & 7;
    float v = acc[0];
    #pragma unroll
    for (int j = 1; j < 8; ++j) {
        if (sel == j) v = acc[j];
    }

    const bool active = (lane < 8) || (lane >= 24);
    const long long eo = base + (lane < 8 ? lane : lane - 16);
    if (active && eo < (long long)E) {
        out[eo] = 1.0f / (1.0f + __expf(-v));   // sigmoid
    }
}

extern "C" void kernel_launch(void* const* d_in, const int* in_sizes, int n_in,
                              void* d_out, int out_size, void* d_ws, size_t ws_size,
                              hipStream_t stream) {
    const float* z          = (const float*)d_in[0];
    const int*   edge_index = (const int*)d_in[1];
    float*       out        = (float*)d_out;

    const int E = out_size;   // 1,250,000 edge scores

    const int grid = (E + EDGES_PER_BLOCK - 1) / EDGES_PER_BLOCK;
    InnerProductDecoder_wmma_kernel<<<grid, 256, 0, stream>>>(z, edge_index, out, E);
}